// Lag2Eul_82300163326236
// MI455X (gfx1250) — compile-verified
//
#include <hip/hip_runtime.h>

// Shapes fixed by the reference: x = (2, 4, 128, 128, 128) f32, out = (2, 1, 128^3) f32
#define P_LOG2   21
#define PSIZE    (1 << P_LOG2)      // 128^3 = 2097152
#define P4SIZE   (PSIZE >> 2)       // float4 count per plane
#define NBATCH   2
#define NBLK     512                // stage-1 blocks per (n,c) slice
#define CHUNK    (PSIZE / NBLK)     // 4096 elements per block
#define DIS_NORM 3.072f             // 6 * 512 / 1000

typedef __attribute__((ext_vector_type(2))) float v2f;
typedef __attribute__((ext_vector_type(4))) float v4f;
typedef __attribute__((ext_vector_type(8))) float v8f;

// Sum 32 lane values of a wave using V_WMMA_F32_16X16X4_F32 with B = ones.
// A: vgpr0 = lane value (A[m][0] for lanes 0-15, A[m][2] for lanes 16-31), vgpr1 = 0.
// D[m][n] = sum_k A[m][k] = v_m + v_{m+16}. Summing the 8 D vgprs per lane gives
// sum of rowsums 0..7 in lanes 0-15 and rowsums 8..15 in lanes 16-31.
__device__ __forceinline__ float wave_sum_wmma(float v) {
    v2f a; a[0] = v;    a[1] = 0.0f;
    v2f b; b[0] = 1.0f; b[1] = 1.0f;
    v8f c = {};
    c = __builtin_amdgcn_wmma_f32_16x16x4_f32(
            /*neg_a=*/false, a, /*neg_b=*/false, b,
            /*c_mod=*/(short)0, c, /*reuse_a=*/false, /*reuse_b=*/false);
    float t = c[0] + c[1] + c[2] + c[3] + c[4] + c[5] + c[6] + c[7];
    return __shfl(t, 0, 32) + __shfl(t, 16, 32);
}

// Deterministic block (256-thread) sum; result valid on thread 0.
__device__ __forceinline__ float block_sum(float v, float* lds) {
    float wsum = wave_sum_wmma(v);          // all 8 waves fully active -> EXEC all 1s
    int lane = threadIdx.x & 31;
    int wid  = threadIdx.x >> 5;
    if (lane == 0) lds[wid] = wsum;
    __syncthreads();
    float total = 0.0f;
    if (threadIdx.x == 0) {
        #pragma unroll
        for (int i = 0; i < 8; ++i) total += lds[i];  // fixed order -> deterministic
    }
    return total;
}

__global__ void zero_out_kernel(v4f* __restrict__ out, int n4) {
    int i = blockIdx.x * blockDim.x + threadIdx.x;
    if (i < n4) out[i] = (v4f){0.f, 0.f, 0.f, 0.f};   // b128 stores, RT (keep mesh in L2)
}

// Stage 1: per-(n,c) slice partial sums. grid = (NBLK, 6), block = 256.
// Non-temporal b128 streaming reads: touched once, keep L2 free for the mesh.
__global__ void reduce1_kernel(const float* __restrict__ x, float* __restrict__ partials) {
    __shared__ float lds[8];
    int slice = blockIdx.y;                 // slice = n*3 + c, c in 0..2
    int n = slice / 3, c = slice % 3;
    const v4f* plane4 = (const v4f*)(x + ((size_t)n * 4 + c) * (size_t)PSIZE);
    int idx = blockIdx.x * (CHUNK / 4) + threadIdx.x;   // float4 index
    float s = 0.0f;
    #pragma unroll
    for (int j = 0; j < CHUNK / 4 / 256; ++j) {         // 4 iterations
        v4f d = __builtin_nontemporal_load(plane4 + idx);
        s += (d[0] + d[1]) + (d[2] + d[3]);
        idx += 256;
    }
    float tot = block_sum(s, lds);
    if (threadIdx.x == 0) partials[slice * NBLK + blockIdx.x] = tot;
}

// Stage 2: fold NBLK partials per slice into a mean. grid = 6 blocks of 256.
__global__ void reduce2_kernel(const float* __restrict__ partials, float* __restrict__ means) {
    __shared__ float lds[8];
    int slice = blockIdx.x;
    const float* p = partials + slice * NBLK;
    float v = p[threadIdx.x] + p[threadIdx.x + 256];
    float tot = block_sum(v, lds);
    if (threadIdx.x == 0) means[slice] = tot * (1.0f / (float)PSIZE);
}

// CIC scatter: 4 particles per thread (b128 NT loads), 8 fp32 hardware atomics per
// particle into the L2-resident mesh (16MB << 192MB L2). Branchless OOB handling:
// per-axis med3 clamp of the target index + cndmask-selected 0.0f deposit value.
// Adding +0.0 is exact (mesh cells are never -0), matching the reference's
// mask-to-zero + clipped-index scatter semantics without any exec-mask branches.
__global__ void scatter_kernel(const float* __restrict__ x,
                               const float* __restrict__ means,
                               float* __restrict__ out) {
    int t = blockIdx.x * blockDim.x + threadIdx.x;   // 0 .. N*P/4-1
    int n = t >> (P_LOG2 - 2);
    int q = t & (P4SIZE - 1);                        // float4 index within slice

    const v4f* xb4 = (const v4f*)(x + (size_t)n * 4 * (size_t)PSIZE);
    v4f dxv = __builtin_nontemporal_load(xb4 + q);
    v4f dyv = __builtin_nontemporal_load(xb4 + P4SIZE + q);
    v4f dzv = __builtin_nontemporal_load(xb4 + 2 * P4SIZE + q);
    v4f vav = __builtin_nontemporal_load(xb4 + 3 * P4SIZE + q);

    float m0 = means[n * 3 + 0];
    float m1 = means[n * 3 + 1];
    float m2 = means[n * 3 + 2];

    float* outn = out + (size_t)n * (size_t)PSIZE;

    int p0 = q << 2;
    int d  = p0 >> 14;             // D index (constant across the 4 particles)
    int h  = (p0 >> 7) & 127;      // H index (constant across the 4 particles)
    int w0 = p0 & 127;             // W index of particle 0 (q*4 stays within a W-row)

    #pragma unroll
    for (int j = 0; j < 4; ++j) {
        float px = (dxv[j] - m0) * DIS_NORM + (float)d + 0.5f;
        float py = (dyv[j] - m1) * DIS_NORM + (float)h + 0.5f;
        float pz = (dzv[j] - m2) * DIS_NORM + (float)(w0 + j) + 0.5f;
        float val = vav[j];

        float fpx = floorf(px), fpy = floorf(py), fpz = floorf(pz);
        int ix = (int)fpx, iy = (int)fpy, iz = (int)fpz;
        float fx = px - fpx, fy = py - fpy, fz = pz - fpz;

        float wx[2] = {1.0f - fx, fx};
        float wy[2] = {1.0f - fy, fy};
        float wz[2] = {1.0f - fz, fz};

        // Per-axis clamped indices and in-bounds flags (v_med3 + cmp, no branches).
        int  xc[2], yc[2], zc[2];
        bool okx[2], oky[2], okz[2];
        #pragma unroll
        for (int a = 0; a < 2; ++a) {
            int vx = ix + a, vy = iy + a, vz = iz + a;
            okx[a] = ((unsigned)vx < 128u);
            oky[a] = ((unsigned)vy < 128u);
            okz[a] = ((unsigned)vz < 128u);
            xc[a] = min(max(vx, 0), 127);
            yc[a] = min(max(vy, 0), 127);
            zc[a] = min(max(vz, 0), 127);
        }

        #pragma unroll
        for (int a = 0; a < 2; ++a) {
            #pragma unroll
            for (int b = 0; b < 2; ++b) {
                float wab  = val * wx[a] * wy[b];
                bool  okab = okx[a] && oky[b];
                int   base = ((xc[a] << 7) + yc[b]) << 7;
                #pragma unroll
                for (int g = 0; g < 2; ++g) {
                    float wv  = (okab && okz[g]) ? wab * wz[g] : 0.0f;  // v_cndmask
                    int   ind = base + zc[g];
                    unsafeAtomicAdd(outn + ind, wv);   // global_atomic_add_f32 (no return)
                }
            }
        }
    }
}

extern "C" void kernel_launch(void* const* d_in, const int* in_sizes, int n_in,
                              void* d_out, int out_size, void* d_ws, size_t ws_size,
                              hipStream_t stream) {
    const float* x   = (const float*)d_in[0];
    float*       out = (float*)d_out;
    float*       ws  = (float*)d_ws;

    float* partials = ws;                 // 6 * NBLK floats
    float* means    = ws + 6 * NBLK;      // 6 floats

    // 1) Zero the output mesh (harness poisons it; must re-zero every call).
    int n4 = out_size / 4;                // float4 count
    zero_out_kernel<<<(n4 + 255) / 256, 256, 0, stream>>>((v4f*)out, n4);

    // 2) Deterministic two-stage mean-displacement reduction (WMMA wave sums).
    dim3 g1(NBLK, 6);
    reduce1_kernel<<<g1, 256, 0, stream>>>(x, partials);
    reduce2_kernel<<<6, 256, 0, stream>>>(partials, means);

    // 3) CIC scatter with hardware fp32 global atomics (mesh resident in 192MB L2).
    int nquads = (NBATCH * PSIZE) / 4;
    scatter_kernel<<<nquads / 256, 256, 0, stream>>>(x, means, out);
}